// SOnEquivalentLayer_3968549782330
// MI455X (gfx1250) — compile-verified
//
#include <hip/hip_runtime.h>
#include <math.h>

typedef __attribute__((ext_vector_type(2))) float v2f;
typedef __attribute__((ext_vector_type(8))) float v8f;

#define BB 2
#define NN 10000
#define MM 48
#define CC 64
#define NBAS 8
#define TOT (BB * NN)            /* 20000 flattened nodes */
#define NODES_PER_BLK 16
#define THREADS 384              /* 12 waves of 32 */
#define AGG_FLOATS (13 * 64 * 16)  /* agg[s][c][node]            = 13312 */
#define WT_FLOATS  (3 * 64 * 64)   /* wT[w][c][o] = W_w[o][c]    = 12288 */
#define SS_FLOATS  (16 * 13)       /* S moments per node         =   208 */
#define OFF1 (TOT * 64)          /* o1 starts after o0 */
#define OFF2 (TOT * 64 * 4)      /* o2 starts after o0+o1 */

__device__ __forceinline__ float silu_f(float z) {
    return z / (1.0f + expf(-z));
}

// One 16x16 output tile chain: D += A(16 rows x 64) * B(64 x 16 cols), K-step 4.
// A[M=node, K=c] = aggS[s][c][node]; B[K=c, N=o] = wT[w][c][o].
template <int NT>
__device__ __forceinline__ void wmma_tiles(const float* aggS, const float* wT,
                                           int sBase, int wSel, int o0,
                                           int lo, int hi, v8f (&acc)[NT]) {
#pragma unroll
    for (int t = 0; t < NT; ++t) {
        v8f c = acc[t];
        const float* ag = aggS + ((sBase + t) << 10);          // [c][node], 64x16
        const float* wp = wT + (wSel << 12) + o0 + lo;         // column o0+lo
#pragma unroll
        for (int kb = 0; kb < 64; kb += 4) {
            int k0 = kb + (hi << 1);
            v2f a, b;
            a[0] = ag[(k0 << 4) + lo];
            a[1] = ag[((k0 + 1) << 4) + lo];
            b[0] = wp[(k0 << 6)];
            b[1] = wp[((k0 + 1) << 6)];
            c = __builtin_amdgcn_wmma_f32_16x16x4_f32(
                    false, a, false, b, (short)0, c, false, false);
        }
        acc[t] = c;
    }
}

__global__ __launch_bounds__(THREADS)
void son_layer_kernel(const float* __restrict__ t0, const float* __restrict__ t1,
                      const float* __restrict__ t2, const float* __restrict__ coord,
                      const int* __restrict__ nbr, const float* __restrict__ mask,
                      const float* __restrict__ wrad, const float* __restrict__ W0,
                      const float* __restrict__ b0v, const float* __restrict__ W1,
                      const float* __restrict__ W2, const float* __restrict__ nlw,
                      const float* __restrict__ nlb, float* __restrict__ out) {
    extern __shared__ float smem[];
    float* aggS = smem;                  // [13][64][16]
    float* wT   = smem + AGG_FLOATS;     // [3][64][64] (c-major, i.e. W^T)
    float* sS   = wT + WT_FLOATS;        // [16][13]

    const int tid    = threadIdx.x;
    const int lane   = tid & 31;
    const int waveId = tid >> 5;
    const int bn0    = blockIdx.x * NODES_PER_BLK;

    // ---- stage transposed weights into LDS (coalesced global reads) ----
    for (int i = tid; i < WT_FLOATS; i += THREADS) {
        int w = i >> 12;
        int rem = i & 4095;
        int o = rem >> 6;
        int c = rem & 63;
        const float* Ws = (w == 0) ? W0 : ((w == 1) ? W1 : W2);
        wT[(w << 12) + (c << 6) + o] = Ws[rem];  // Ws[o*64+c]
    }

    // ---- neighbor-sum moments: one wave per node, shuffle reduction ----
    float wr[NBAS];
#pragma unroll
    for (int k = 0; k < NBAS; ++k) wr[k] = wrad[k];

    for (int nl = waveId; nl < NODES_PER_BLK; nl += (THREADS / 32)) {
        int g = bn0 + nl;
        float p[13];
#pragma unroll
        for (int k = 0; k < 13; ++k) p[k] = 0.0f;
        if (g < TOT) {
            int bb = g / NN;
            float cix = coord[g * 3 + 0];
            float ciy = coord[g * 3 + 1];
            float ciz = coord[g * 3 + 2];
            for (int m = lane; m < MM; m += 32) {
                int j = nbr[g * MM + m];
                int cj = (bb * NN + j) * 3;
                float rx = coord[cj + 0] - cix;
                float ry = coord[cj + 1] - ciy;
                float rz = coord[cj + 2] - ciz;
                float d = sqrtf(rx * rx + ry * ry + rz * rz + 1e-12f);
                float x = 0.4f * d - 1.0f;                 // 2*d/RCUT - 1
                float Tp = 1.0f, Tc = x;
                float fnm = wr[0] + wr[1] * x;
#pragma unroll
                for (int k = 2; k < NBAS; ++k) {
                    float Tn = 2.0f * x * Tc - Tp;
                    fnm += wr[k] * Tn;
                    Tp = Tc; Tc = Tn;
                }
                float fc = (d < 5.0f) ? 0.5f * (cosf(0.6283185307179586f * d) + 1.0f)
                                      : 0.0f;
                float fn = fnm * fc * mask[g * MM + m];
                p[0]  += fn;
                p[1]  += fn * rx;  p[2]  += fn * ry;  p[3]  += fn * rz;
                p[4]  += fn * rx * rx;  p[5]  += fn * rx * ry;  p[6]  += fn * rx * rz;
                p[7]  += fn * ry * rx;  p[8]  += fn * ry * ry;  p[9]  += fn * ry * rz;
                p[10] += fn * rz * rx;  p[11] += fn * rz * ry;  p[12] += fn * rz * rz;
            }
        }
#pragma unroll
        for (int off = 16; off > 0; off >>= 1) {
#pragma unroll
            for (int k = 0; k < 13; ++k) p[k] += __shfl_xor(p[k], off, 32);
        }
        if (lane == 0) {
#pragma unroll
            for (int k = 0; k < 13; ++k) sS[nl * 13 + k] = p[k];
        }
    }
    __syncthreads();

    // ---- build agg[13][64][16] in LDS: per (node, channel) pair ----
    for (int pr = tid; pr < 1024; pr += THREADS) {
        int nl = pr >> 6;
        int c  = pr & 63;
        int g  = bn0 + nl;
        float a0 = 0.0f, a1[3] = {0, 0, 0};
        float a2[9] = {0, 0, 0, 0, 0, 0, 0, 0, 0};
        if (g < TOT) {
            int base = g * 64 + c;
            a0 = t0[base];
#pragma unroll
            for (int q = 0; q < 3; ++q) a1[q] = t1[base * 3 + q];
#pragma unroll
            for (int e = 0; e < 9; ++e) a2[e] = t2[base * 9 + e];
        }
        float s0 = sS[nl * 13 + 0];
        float s1[3], s2[9];
#pragma unroll
        for (int q = 0; q < 3; ++q) s1[q] = sS[nl * 13 + 1 + q];
#pragma unroll
        for (int e = 0; e < 9; ++e) s2[e] = sS[nl * 13 + 4 + e];

        // out-way 0
        float r0 = a0 * s0;
#pragma unroll
        for (int q = 0; q < 3; ++q) r0 += a1[q] * s1[q];
#pragma unroll
        for (int e = 0; e < 9; ++e) r0 += a2[e] * s2[e];
        aggS[(0 << 10) + (c << 4) + nl] = r0;

        // out-way 1
#pragma unroll
        for (int q = 0; q < 3; ++q) {
            float v = a0 * s1[q] + a1[q] * s0;
#pragma unroll
            for (int u = 0; u < 3; ++u)
                v += a1[u] * s2[u * 3 + q] + a2[q * 3 + u] * s1[u];
            aggS[((1 + q) << 10) + (c << 4) + nl] = v;
        }

        // out-way 2
#pragma unroll
        for (int q = 0; q < 3; ++q) {
#pragma unroll
            for (int r = 0; r < 3; ++r) {
                float v = a0 * s2[q * 3 + r] + a1[q] * s1[r] + a2[q * 3 + r] * s0;
#pragma unroll
                for (int u = 0; u < 3; ++u) v += a2[q * 3 + u] * s2[u * 3 + r];
                aggS[((4 + q * 3 + r) << 10) + (c << 4) + nl] = v;
            }
        }
    }
    __syncthreads();

    // ---- WMMA channel mix + gated nonlinearity + store ----
    const int lo = lane & 15;
    const int hi = lane >> 4;
    const int o0 = (waveId & 3) << 4;
    const int wClass = (waveId < 4) ? 2 : ((waveId < 8) ? 1 : 0);
    const v8f vzero = {0, 0, 0, 0, 0, 0, 0, 0};

    if (wClass == 2) {
        v8f acc[9];
#pragma unroll
        for (int t = 0; t < 9; ++t) acc[t] = vzero;
        wmma_tiles<9>(aggS, wT, 4, 2, o0, lo, hi, acc);
        float gw = nlw[2], gb = nlb[2];
#pragma unroll
        for (int jr = 0; jr < 8; ++jr) {
            int nl = jr + (hi << 3);
            int g = bn0 + nl;
            if (g < TOT) {
                float y[9], ss = 0.0f;
#pragma unroll
                for (int e = 0; e < 9; ++e) { y[e] = acc[e][jr]; ss += y[e] * y[e]; }
                float sc = silu_f(gw * sqrtf(ss) + gb);
                int ob = OFF2 + (g * 64 + o0 + lo) * 9;
#pragma unroll
                for (int e = 0; e < 9; ++e) out[ob + e] = sc * y[e];
            }
        }
    } else if (wClass == 1) {
        v8f acc[3];
#pragma unroll
        for (int t = 0; t < 3; ++t) acc[t] = vzero;
        wmma_tiles<3>(aggS, wT, 1, 1, o0, lo, hi, acc);
        float gw = nlw[1], gb = nlb[1];
#pragma unroll
        for (int jr = 0; jr < 8; ++jr) {
            int nl = jr + (hi << 3);
            int g = bn0 + nl;
            if (g < TOT) {
                float y[3], ss = 0.0f;
#pragma unroll
                for (int e = 0; e < 3; ++e) { y[e] = acc[e][jr]; ss += y[e] * y[e]; }
                float sc = silu_f(gw * sqrtf(ss) + gb);
                int ob = OFF1 + (g * 64 + o0 + lo) * 3;
#pragma unroll
                for (int e = 0; e < 3; ++e) out[ob + e] = sc * y[e];
            }
        }
    } else {
        v8f acc[1];
        acc[0] = vzero;
        wmma_tiles<1>(aggS, wT, 0, 0, o0, lo, hi, acc);
        float gw = nlw[0], gb = nlb[0];
        float bo = b0v[o0 + lo];
#pragma unroll
        for (int jr = 0; jr < 8; ++jr) {
            int nl = jr + (hi << 3);
            int g = bn0 + nl;
            if (g < TOT) {
                float z = gw * (acc[0][jr] + bo) + gb;
                out[g * 64 + o0 + lo] = silu_f(z);
            }
        }
    }
}

extern "C" void kernel_launch(void* const* d_in, const int* in_sizes, int n_in,
                              void* d_out, int out_size, void* d_ws, size_t ws_size,
                              hipStream_t stream) {
    const float* t0    = (const float*)d_in[0];
    const float* t1    = (const float*)d_in[1];
    const float* t2    = (const float*)d_in[2];
    const float* coord = (const float*)d_in[3];
    const int*   nbr   = (const int*)d_in[4];
    const float* mask  = (const float*)d_in[5];
    const float* wrad  = (const float*)d_in[6];
    const float* W0    = (const float*)d_in[7];
    const float* b0    = (const float*)d_in[8];
    const float* W1    = (const float*)d_in[9];
    const float* W2    = (const float*)d_in[10];
    const float* nlw   = (const float*)d_in[11];
    const float* nlb   = (const float*)d_in[12];
    float* out = (float*)d_out;

    int blocks = (TOT + NODES_PER_BLK - 1) / NODES_PER_BLK;  // 1250
    size_t shmem = (size_t)(AGG_FLOATS + WT_FLOATS + SS_FLOATS) * sizeof(float);
    son_layer_kernel<<<dim3(blocks), dim3(THREADS), shmem, stream>>>(
        t0, t1, t2, coord, nbr, mask, wrad, W0, b0, W1, W2, nlw, nlb, out);
}